// GaussianModel_34308198761191
// MI455X (gfx1250) — compile-verified
//
#include <hip/hip_runtime.h>

typedef __attribute__((ext_vector_type(2))) float v2f;
typedef __attribute__((ext_vector_type(8))) float v8f;

#define VOLD 256
#define W 32

__global__ __launch_bounds__(256) void zero_volume(float* __restrict__ out, int n4) {
  int i = blockIdx.x * blockDim.x + threadIdx.x;
  if (i < n4) {
    float4 z = {0.f, 0.f, 0.f, 0.f};
    ((float4*)out)[i] = z;
  }
}

__global__ __launch_bounds__(256) void gauss_splat(const float* __restrict__ centers,
                                                   const float* __restrict__ sigmas,
                                                   const float* __restrict__ intens,
                                                   float* __restrict__ vol) {
  __shared__ float gAx[3][W];   // masked 1-D gaussians: x, y, z (intensity folded into z)
  __shared__ int   sstart[3];
  __shared__ float P[W][W];     // outer product, stored TRANSPOSED: P[y][x]

  const int g    = blockIdx.x;
  const int t    = threadIdx.x;
  const int wave = t >> 5;
  const int lane = t & 31;
  const float scale = (float)(VOLD - 1);

  // ---- phase 0: per-axis windows (threads 0..95) ----
  if (t < 96) {
    const int axis = t >> 5;        // 0,1,2
    const int off  = t & 31;
    const float cN   = centers[g * 3 + axis];      // normalized center
    const float sig  = sigmas[g];
    const float c    = cN * scale;                 // voxel-space center
    const float cut  = 3.0f * sig * scale;
    const float minf = fmaxf(c - cut, 0.0f);
    const float maxf = fminf(c + cut, scale);
    const int   min_i = (int)floorf(minf);
    const int   max_i = min((int)floorf(maxf) + 1, VOLD);
    const int   start = min(max(min_i, 0), VOLD - W);
    const int   idx   = start + off;
    const bool  valid = (idx >= min_i) && (idx < max_i);
    const float pos   = (float)idx / scale;        // normalized grid coord
    const float d     = pos - cN;
    const float inv2s2 = 0.5f / (sig * sig);
    float val = valid ? __expf(-d * d * inv2s2) : 0.0f;
    if (axis == 2) val *= intens[g];               // fold intensity into z factor
    gAx[axis][off] = val;
    if (off == 0) sstart[axis] = start;
  }
  __syncthreads();

  // ---- phase 1: waves 0..3 each compute one 16x16 tile of the 32x32 outer
  // product via V_WMMA_F32_16X16X4_F32 (K=0 carries the data, K=1..3 zero).
  // Branch is wave-uniform -> EXEC all-1s inside, as WMMA requires.
  if (wave < 4) {
    const int tx = wave & 1;
    const int ty = wave >> 1;
    const bool lo = lane < 16;
    const int  l16 = lane & 15;

    // A: 16x4 f32. VGPR0 = K=0 (lanes 0-15) / K=2 (lanes 16-31); VGPR1 = K=1/K=3.
    v2f a;
    a.x = lo ? gAx[0][tx * 16 + l16] : 0.0f;
    a.y = 0.0f;
    // B: 4x16 f32, symmetric striping: VGPR0 lanes 0-15 carry K=0 row.
    v2f b;
    b.x = lo ? gAx[1][ty * 16 + l16] : 0.0f;
    b.y = 0.0f;

    v8f acc = {};
    acc = __builtin_amdgcn_wmma_f32_16x16x4_f32(
        /*neg_a=*/false, a, /*neg_b=*/false, b,
        /*c_mod=*/(short)0, acc, /*reuse_a=*/false, /*reuse_b=*/false);

    // D layout: VGPR r -> M = r (lanes 0-15) / M = r+8 (lanes 16-31), N = lane%16.
    // Store transposed P[y][x]: per lane the 8 values land in one contiguous,
    // 32B-aligned run of 8 floats -> merges into ds_store_b128s.
    const int y  = ty * 16 + l16;
    const int x0 = tx * 16 + (lo ? 0 : 8);
    #pragma unroll
    for (int r = 0; r < 8; ++r) {
      P[y][x0 + r] = acc[r];
    }
  }
  __syncthreads();

  // ---- phase 2: scatter-add. lane = z offset (contiguous in memory).
  // For each x, this wave owns rows y in {wave, wave+8, wave+16, wave+24}.
  // Integer indexing keeps atomics in SADDR + 32-bit voffset scale_offset form.
  // Single v!=0 guard: p==0 or gz==0 (masked voxels) both imply v==0 exactly,
  // and adding 0.0 is a no-op, so one compare gives the identical skip set.
  const int sx = sstart[0];
  const int sy = sstart[1];
  const int sz = sstart[2];
  const float gz = gAx[2][lane];
  const int y0 = wave;

  for (int x = 0; x < W; ++x) {
    const float p0 = P[y0][x];
    const float p1 = P[y0 + 8][x];
    const float p2 = P[y0 + 16][x];
    const float p3 = P[y0 + 24][x];
    // 32-bit element index of (sx+x, sy+y0, sz+lane)
    const int rowbase = (((sx + x) * VOLD) + sy + y0) * VOLD + sz + lane;
    const float v0 = p0 * gz;
    const float v1 = p1 * gz;
    const float v2 = p2 * gz;
    const float v3 = p3 * gz;
    if (v0 != 0.0f) atomicAdd(&vol[rowbase], v0);
    if (v1 != 0.0f) atomicAdd(&vol[rowbase + 8 * VOLD], v1);
    if (v2 != 0.0f) atomicAdd(&vol[rowbase + 16 * VOLD], v2);
    if (v3 != 0.0f) atomicAdd(&vol[rowbase + 24 * VOLD], v3);
  }
}

extern "C" void kernel_launch(void* const* d_in, const int* in_sizes, int n_in,
                              void* d_out, int out_size, void* d_ws, size_t ws_size,
                              hipStream_t stream) {
  const float* centers = (const float*)d_in[0];
  const float* sigmas  = (const float*)d_in[1];
  const float* intens  = (const float*)d_in[2];
  float* out = (float*)d_out;

  // zero the volume every call (graph replays don't re-poison/restore)
  const int n4 = out_size / 4;
  zero_volume<<<(n4 + 255) / 256, 256, 0, stream>>>(out, n4);

  const int n_gauss = in_sizes[1];   // sigmas: one per gaussian
  gauss_splat<<<n_gauss, 256, 0, stream>>>(centers, sigmas, intens, out);
}